// DBSCANLayer_39737037422641
// MI455X (gfx1250) — compile-verified
//
#include <hip/hip_runtime.h>
#include <hip/hip_bf16.h>
#include <stdint.h>

// DBSCAN on MI455X (gfx1250, wave32).
// Pipeline:
//  1) norms:   sq[i] = |x_i|^2
//  2) adj:     FP32 WMMA (v_wmma_f32_16x16x4_f32) Gram tiles -> threshold in
//              registers -> ballot-packed 8192x8192 bit adjacency (8 MB, L2-resident)
//  3) degree:  popcount rows -> core flags, core bitmask, initial labels
//  4) prop x16: min-label propagation over core-core graph (LDS label cache)
//  5) border:  non-core points take min label of adjacent core points
//  6) finalize: single-block scan renumbers roots 0..k-1, noise -> -1

#define N      8192
#define D      128
#define WORDS  (N / 32)      // 256 words per adjacency row
#define EPS2   81.0f
#define MINPTS 5
#define BIG    N

typedef __attribute__((ext_vector_type(2))) float v2f;
typedef __attribute__((ext_vector_type(8))) float v8f;

// ---------------------------------------------------------------- norms
__global__ void norms_kernel(const float* __restrict__ x, float* __restrict__ sq) {
    int i = blockIdx.x * 256 + threadIdx.x;
    const float4* r = (const float4*)(x + (size_t)i * D);
    float s = 0.0f;
#pragma unroll 8
    for (int j = 0; j < D / 4; ++j) {
        float4 v = r[j];
        s += v.x * v.x + v.y * v.y + v.z * v.z + v.w * v.w;
    }
    sq[i] = s;
}

// ---------------------------------------------------------------- adjacency
// Each wave: one 16(M) x 32(N) tile of the Gram matrix via FP32 WMMA.
// A operand (16x4 f32): lane L<16 holds x[i0+L][k..k+1]; lane L>=16 holds
// x[i0+L-16][k+2..k+3]. B operand (4x16) mirrors it with the column base.
__global__ void adj_kernel(const float* __restrict__ x,
                           const float* __restrict__ sq,
                           uint32_t* __restrict__ adj) {
    int lane  = threadIdx.x & 31;
    int wid   = threadIdx.x >> 5;
    int wcol  = blockIdx.x;                 // word column: 32 N-columns
    int mtile = blockIdx.y * 4 + wid;       // 0..511
    int i0 = mtile * 16;
    int j0 = wcol * 32;

    int hi = lane >> 4;                     // 0: lanes 0-15, 1: lanes 16-31
    int l15 = lane & 15;

    const float* xa  = x + (size_t)(i0 + l15) * D + 2 * hi;
    const float* xb0 = x + (size_t)(j0 + l15) * D + 2 * hi;
    const float* xb1 = xb0 + 16 * D;

    v8f acc0 = {}, acc1 = {};
#pragma unroll 4
    for (int k = 0; k < D; k += 4) {
        v2f a  = *(const v2f*)(xa + k);
        v2f b0 = *(const v2f*)(xb0 + k);
        v2f b1 = *(const v2f*)(xb1 + k);
        acc0 = __builtin_amdgcn_wmma_f32_16x16x4_f32(false, a, false, b0,
                                                     (short)0, acc0, false, false);
        acc1 = __builtin_amdgcn_wmma_f32_16x16x4_f32(false, a, false, b1,
                                                     (short)0, acc1, false, false);
    }

    float sqj0 = sq[j0 + l15];
    float sqj1 = sq[j0 + 16 + l15];

    // C layout: VGPR v holds (M=v, lanes 0-15) and (M=v+8, lanes 16-31).
    uint32_t bal0[8], bal1[8];
#pragma unroll
    for (int v = 0; v < 8; ++v) {
        float sqi = sq[i0 + v + (hi ? 8 : 0)];
        bal0[v] = __builtin_amdgcn_ballot_w32(sqi + sqj0 - 2.0f * acc0[v] <= EPS2);
        bal1[v] = __builtin_amdgcn_ballot_w32(sqi + sqj1 - 2.0f * acc1[v] <= EPS2);
    }

    // Ballots are wave-uniform; lane m (<16) assembles the word for row i0+m.
    uint32_t w0 = 0, w1 = 0;
#pragma unroll
    for (int v = 0; v < 8; ++v) {
        if ((lane & 7) == v) { w0 = bal0[v]; w1 = bal1[v]; }
    }
    uint32_t word = (lane < 8)
        ? ((w0 & 0xFFFFu) | ((w1 & 0xFFFFu) << 16))   // rows 0..7
        : ((w0 >> 16)     | (w1 & 0xFFFF0000u));      // rows 8..15
    if (lane < 16)
        adj[(size_t)(i0 + lane) * WORDS + wcol] = word;
}

// ---------------------------------------------------------------- degree/core
__global__ void degree_kernel(const uint32_t* __restrict__ adj,
                              int* __restrict__ core,
                              uint32_t* __restrict__ corebits,
                              int* __restrict__ lbl0) {
    int lane = threadIdx.x & 31;
    int row  = blockIdx.x * 8 + (threadIdx.x >> 5);
    const uint32_t* arow = adj + (size_t)row * WORDS;
    int cnt = 0;
#pragma unroll
    for (int w = lane; w < WORDS; w += 32) cnt += __popc(arow[w]);
#pragma unroll
    for (int off = 16; off; off >>= 1) cnt += __shfl_xor(cnt, off, 32);
    if (lane == 0) {
        int c = (cnt >= MINPTS) ? 1 : 0;
        core[row] = c;
        lbl0[row] = c ? row : BIG;
        if (c) atomicOr(&corebits[row >> 5], 1u << (row & 31));
    }
}

// ---------------------------------------------------------------- propagation
__global__ void prop_kernel(const uint32_t* __restrict__ adj,
                            const uint32_t* __restrict__ corebits,
                            const int* __restrict__ core,
                            const int* __restrict__ lin,
                            int* __restrict__ lout) {
    __shared__ int slbl[N];
    for (int i = threadIdx.x; i < N; i += 256) slbl[i] = lin[i];
    __syncthreads();

    int lane = threadIdx.x & 31;
    int row  = blockIdx.x * 8 + (threadIdx.x >> 5);
    int best = BIG;
    if (core[row]) {
        best = slbl[row];
        const uint32_t* arow = adj + (size_t)row * WORDS;
        for (int w = lane; w < WORDS; w += 32) {
            uint32_t m = arow[w] & corebits[w];
            while (m) {
                int b = __builtin_ctz(m);
                m &= m - 1;
                best = min(best, slbl[w * 32 + b]);
            }
        }
    }
#pragma unroll
    for (int off = 16; off; off >>= 1) best = min(best, __shfl_xor(best, off, 32));
    if (lane == 0) lout[row] = best;
}

// ---------------------------------------------------------------- border
__global__ void border_kernel(const uint32_t* __restrict__ adj,
                              const uint32_t* __restrict__ corebits,
                              const int* __restrict__ core,
                              const int* __restrict__ lbl,
                              int* __restrict__ raw) {
    __shared__ int slbl[N];
    for (int i = threadIdx.x; i < N; i += 256) slbl[i] = lbl[i];
    __syncthreads();

    int lane = threadIdx.x & 31;
    int row  = blockIdx.x * 8 + (threadIdx.x >> 5);
    int best = BIG;
    const uint32_t* arow = adj + (size_t)row * WORDS;
    for (int w = lane; w < WORDS; w += 32) {
        uint32_t m = arow[w] & corebits[w];   // only core neighbors
        while (m) {
            int b = __builtin_ctz(m);
            m &= m - 1;
            best = min(best, slbl[w * 32 + b]);
        }
    }
#pragma unroll
    for (int off = 16; off; off >>= 1) best = min(best, __shfl_xor(best, off, 32));
    if (lane == 0) raw[row] = core[row] ? slbl[row] : best;
}

// ---------------------------------------------------------------- finalize
// Single block, 1024 threads: is_root -> inclusive scan -> rank gather.
__global__ void finalize_kernel(const int* __restrict__ raw,
                                const int* __restrict__ core,
                                int* __restrict__ out) {
    __shared__ int rank[N];
    __shared__ int tsum[1024];
    int t = threadIdx.x;
    int base = t * 8;

    int local[8];
    int s = 0;
#pragma unroll
    for (int j = 0; j < 8; ++j) {
        int i = base + j;
        int r = (raw[i] == i && core[i]) ? 1 : 0;
        s += r;
        local[j] = s;                       // inclusive within thread
    }
    tsum[t] = s;
    __syncthreads();
    for (int off = 1; off < 1024; off <<= 1) {
        int v = (t >= off) ? tsum[t - off] : 0;
        __syncthreads();
        tsum[t] += v;
        __syncthreads();
    }
    int excl = tsum[t] - s;
#pragma unroll
    for (int j = 0; j < 8; ++j) rank[base + j] = excl + local[j] - 1;
    __syncthreads();
#pragma unroll
    for (int j = 0; j < 8; ++j) {
        int i = base + j;
        int r = raw[i];
        out[i] = (r < BIG) ? rank[min(r, N - 1)] : -1;
    }
}

// ---------------------------------------------------------------- launch
extern "C" void kernel_launch(void* const* d_in, const int* in_sizes, int n_in,
                              void* d_out, int out_size, void* d_ws, size_t ws_size,
                              hipStream_t stream) {
    const float* x = (const float*)d_in[0];

    uint8_t* ws = (uint8_t*)d_ws;
    size_t off = 0;
    uint32_t* adj      = (uint32_t*)(ws + off); off += (size_t)N * WORDS * 4;  // 8 MB
    float*    sq       = (float*)(ws + off);    off += N * 4;
    int*      core     = (int*)(ws + off);      off += N * 4;
    uint32_t* corebits = (uint32_t*)(ws + off); off += WORDS * 4;
    int*      lblA     = (int*)(ws + off);      off += N * 4;
    int*      lblB     = (int*)(ws + off);      off += N * 4;
    int*      raw      = (int*)(ws + off);      off += N * 4;

    hipMemsetAsync(corebits, 0, WORDS * 4, stream);

    norms_kernel<<<N / 256, 256, 0, stream>>>(x, sq);

    // 512 M-tiles x 256 word-columns; 4 waves (M-tiles) per 128-thread block.
    adj_kernel<<<dim3(WORDS, 512 / 4), 128, 0, stream>>>(x, sq, adj);

    degree_kernel<<<N / 8, 256, 0, stream>>>(adj, core, corebits, lblA);

    for (int it = 0; it < 16; ++it) {
        const int* lin = (it & 1) ? lblB : lblA;
        int*       lo  = (it & 1) ? lblA : lblB;
        prop_kernel<<<N / 8, 256, 0, stream>>>(adj, corebits, core, lin, lo);
    }
    // 16 iterations: final labels back in lblA.

    border_kernel<<<N / 8, 256, 0, stream>>>(adj, corebits, core, lblA, raw);

    finalize_kernel<<<1, 1024, 0, stream>>>(raw, core, (int*)d_out);
}